// NTMCell_41497974014488
// MI455X (gfx1250) — compile-verified
//
#include <hip/hip_runtime.h>
#include <hip/hip_bf16.h>
#include <math.h>

#define NUM_HEADS 4
#define MSZ 64
#define BSZ 128
#define IN_DIM 256
#define STATE 512
#define OUT_DIM 256
#define N_ADDR 2048
#define PER_HEAD 198
#define UPDATE_SIZE 792
#define CTRL_IN 1024
#define GATE_IN 768
#define EPSF 1e-8f

typedef __attribute__((ext_vector_type(2))) float v2f;
typedef __attribute__((ext_vector_type(8))) float v8f;

__device__ __forceinline__ float sigmoidf_(float x) { return 1.0f / (1.0f + expf(-x)); }
__device__ __forceinline__ float softplusf_(float x) { return (x > 20.0f) ? x : log1pf(expf(x)); }

// ---------------- K1: gates f (Bx4), h (Bx2) ----------------
__global__ void gates_kernel(const float* __restrict__ tin, const float* __restrict__ tst,
                             const float* __restrict__ Wf, const float* __restrict__ bf,
                             const float* __restrict__ Wh, const float* __restrict__ bh,
                             float* __restrict__ f, float* __restrict__ h) {
    int b = blockIdx.x;
    int wave = threadIdx.x >> 5;   // 0..5
    int lane = threadIdx.x & 31;
    float acc = 0.0f;
    for (int k = lane; k < GATE_IN; k += 32) {
        float x = (k < STATE) ? tst[b * STATE + k] : tin[b * IN_DIM + (k - STATE)];
        float w = (wave < 4) ? Wf[k * NUM_HEADS + wave] : Wh[k * 2 + (wave - 4)];
        acc += x * w;
    }
    for (int off = 16; off > 0; off >>= 1) acc += __shfl_xor(acc, off, 32);
    if (lane == 0) {
        float bv = (wave < 4) ? bf[wave] : bh[wave - 4];
        float v = sigmoidf_(acc + bv);
        if (wave < 4) f[b * 4 + wave] = v;
        else          h[b * 2 + (wave - 4)] = v;
    }
}

// ---------------- K2: wt_dyn + wt_mix (elementwise over B*4*2048) ----------------
__global__ void wtmix_kernel(const float* __restrict__ wt, const float* __restrict__ wtd,
                             const float* __restrict__ f, const float* __restrict__ h,
                             float* __restrict__ wt_mix, float* __restrict__ wt_dyn) {
    int idx = blockIdx.x * blockDim.x + threadIdx.x;
    if (idx >= BSZ * NUM_HEADS * N_ADDR) return;
    int b = idx / (NUM_HEADS * N_ADDR);
    int rem = idx - b * (NUM_HEADS * N_ADDR);
    float f0 = f[b * 4 + 0], f1 = f[b * 4 + 1], h0 = h[b * 2 + 0];
    float w = wt[idx], wd = wtd[idx];
    float wdyn = (1.0f - h0) * w + h0 * wd;
    float w0 = (rem == 0) ? 1.0f : 0.0f;
    float wmix = (1.0f - f0) * (1.0f - f1) * w0 + (1.0f - f0) * f1 * w + f0 * (1.0f - f1) * wdyn;
    wt_dyn[idx] = wdyn;
    wt_mix[idx] = wmix;
}

// ---------------- K3: ctrl_in[0:768] = [input, state] ----------------
__global__ void concat_kernel(const float* __restrict__ tin, const float* __restrict__ tst,
                              float* __restrict__ ctrl) {
    int idx = blockIdx.x * blockDim.x + threadIdx.x;
    if (idx >= BSZ * (IN_DIM + STATE)) return;
    int b = idx / GATE_IN, k = idx - b * GATE_IN;
    ctrl[b * CTRL_IN + k] = (k < IN_DIM) ? tin[b * IN_DIM + k] : tst[b * STATE + (k - IN_DIM)];
}

// ---------------- K4a: partial read einsum over 256-wide n chunks ----------------
// grid = B*8 blocks, 256 threads (thread = head*64 + m)
__global__ void read_partial_kernel(const float* __restrict__ wt_mix,
                                    const float* __restrict__ mem,
                                    float* __restrict__ partial) {
    int b = blockIdx.x >> 3;
    int chunk = blockIdx.x & 7;
    int hh = threadIdx.x >> 6;
    int m = threadIdx.x & 63;
    const float* wm = wt_mix + ((size_t)b * NUM_HEADS + hh) * N_ADDR + chunk * 256;
    const float* mb = mem + ((size_t)b * N_ADDR + chunk * 256) * MSZ + m;
    float acc = 0.0f;
    for (int n = 0; n < 256; ++n) acc += wm[n] * mb[(size_t)n * MSZ];
    partial[((size_t)(b * 8 + chunk)) * 256 + threadIdx.x] = acc;
}

// ---------------- K4b: reduce 8 chunks -> ctrl_in[768:1024] ----------------
__global__ void read_reduce_kernel(const float* __restrict__ partial, float* __restrict__ ctrl) {
    int idx = blockIdx.x * blockDim.x + threadIdx.x;   // B*256
    if (idx >= BSZ * 256) return;
    int b = idx >> 8;
    int hm = idx & 255;
    float acc = 0.0f;
#pragma unroll
    for (int c = 0; c < 8; ++c) acc += partial[((size_t)(b * 8 + c)) * 256 + hm];
    ctrl[b * CTRL_IN + GATE_IN + hm] = acc;
}

// ---------------- K5: fp32 WMMA GEMM: C[M,N] = act(A[M,K] @ B[K,N] + bias) ----------------
// One wave computes one 16x16 tile using V_WMMA_F32_16X16X4_F32.
__global__ void wmma_gemm_f32(const float* __restrict__ A, const float* __restrict__ Bw,
                              const float* __restrict__ bias, float* __restrict__ C,
                              int Md, int Nd, int Kd, int act, int totalTiles) {
    int wave = (blockIdx.x * blockDim.x + threadIdx.x) >> 5;
    if (wave >= totalTiles) return;
    int lane = threadIdx.x & 31;
    int tilesM = Md >> 4;
    int tm = wave % tilesM;
    int tn = wave / tilesM;

    int row_a = tm * 16 + (lane & 15);          // A: M = lane%16
    int kp = (lane >> 4) * 2;                   // K pair offset per half-wave
    int col_b = tn * 16 + (lane & 15);          // B: N = lane%16
    bool bvalid = (col_b < Nd);
    int col_c = bvalid ? col_b : (Nd - 1);      // clamp: garbage cols never stored

    const float* aptr = A + (size_t)row_a * Kd + kp;
    const float* bptr = Bw + (size_t)kp * Nd + col_c;
    const size_t bstep = (size_t)4 * Nd;

    v8f acc = {};
#pragma unroll 4
    for (int k0 = 0; k0 < Kd; k0 += 4) {
        v2f a = *(const v2f*)aptr;
        v2f bb = {bptr[0], bptr[Nd]};
        acc = __builtin_amdgcn_wmma_f32_16x16x4_f32(false, a, false, bb,
                                                    (short)0, acc, false, false);
        aptr += 4;
        bptr += bstep;
    }

    if (bvalid) {
        float bv = bias[col_b];
        int rbase = tm * 16 + ((lane >> 4) * 8);   // D vgpr r: rows r / r+8 per half-wave
#pragma unroll
        for (int r = 0; r < 8; ++r) {
            float v = acc[r] + bv;
            if (act) v = sigmoidf_(v);
            C[(size_t)(rbase + r) * Nd + col_b] = v;
        }
    }
}

// ---------------- K6: interface split (per b,h) ----------------
__global__ void iface_kernel(const float* __restrict__ upd,
                             float* __restrict__ erase, float* __restrict__ addv,
                             float* __restrict__ key, float* __restrict__ shift,
                             float* __restrict__ gamma, float* __restrict__ beta,
                             float* __restrict__ gq, float* __restrict__ kn) {
    int bh = blockIdx.x;                       // b*4 + h
    int m = threadIdx.x;                       // 0..63
    const float* u = upd + (size_t)(bh >> 2) * UPDATE_SIZE + (bh & 3) * PER_HEAD;
    erase[bh * MSZ + m] = sigmoidf_(u[m]);
    addv[bh * MSZ + m]  = u[MSZ + m];
    key[bh * MSZ + m]   = u[132 + m];
    if (m == 0) {
        float s0 = u[128], s1 = u[129], s2 = u[130];
        float mx = fmaxf(s0, fmaxf(s1, s2));
        float e0 = expf(s0 - mx), e1 = expf(s1 - mx), e2 = expf(s2 - mx);
        float inv = 1.0f / (e0 + e1 + e2);
        shift[bh * 3 + 0] = e0 * inv;
        shift[bh * 3 + 1] = e1 * inv;
        shift[bh * 3 + 2] = e2 * inv;
        gamma[bh] = 1.0f + softplusf_(u[131]);
        beta[bh]  = softplusf_(u[196]);
        gq[bh]    = sigmoidf_(u[197]);
        float s = 0.0f;
        for (int i = 0; i < MSZ; ++i) { float kv = u[132 + i]; s += kv * kv; }
        kn[bh] = sqrtf(s);
    }
}

// ---------------- K7a: cosine scores, row norms fused (grid B*8, 256 thr) ----------------
__global__ void score_kernel(const float* __restrict__ mem, const float* __restrict__ keyA,
                             const float* __restrict__ betaA, const float* __restrict__ knA,
                             float* __restrict__ score) {
    __shared__ float keys[NUM_HEADS * MSZ];
    int b = blockIdx.x >> 3;
    int chunk = blockIdx.x & 7;
    int t = threadIdx.x;
    keys[t] = keyA[b * NUM_HEADS * MSZ + t];
    __syncthreads();

    int n = chunk * 256 + t;
    const float4* row = (const float4*)(mem + ((size_t)b * N_ADDR + n) * MSZ);
    float d[NUM_HEADS] = {0, 0, 0, 0};
    float nn = 0.0f;
#pragma unroll
    for (int q = 0; q < 16; ++q) {
        float4 v = row[q];
        nn += v.x * v.x + v.y * v.y + v.z * v.z + v.w * v.w;
#pragma unroll
        for (int hh = 0; hh < NUM_HEADS; ++hh) {
            const float* kk = keys + hh * MSZ + q * 4;
            d[hh] += v.x * kk[0] + v.y * kk[1] + v.z * kk[2] + v.w * kk[3];
        }
    }
    float mn = sqrtf(nn);
#pragma unroll
    for (int hh = 0; hh < NUM_HEADS; ++hh) {
        int bh = b * NUM_HEADS + hh;
        score[((size_t)bh) * N_ADDR + n] = betaA[bh] * d[hh] / (knA[bh] * mn + EPSF);
    }
}

// ---------------- K7b: softmax -> gate -> circular conv -> pow -> normalize ----------------
// One block per batch element; touches only the 4MB score buffer + wt_mix.
__global__ void address_kernel(const float* __restrict__ score, const float* __restrict__ wt_mix,
                               const float* __restrict__ shiftA, const float* __restrict__ gammaA,
                               const float* __restrict__ gA, float* __restrict__ wnew) {
    __shared__ float sbuf[NUM_HEADS * N_ADDR];   // 32 KB
    __shared__ float red[NUM_HEADS * 256];       // 4 KB
    int b = blockIdx.x;
    int t = threadIdx.x;                          // 0..255

    float gv[NUM_HEADS], gm[NUM_HEADS], sh0[NUM_HEADS], sh1[NUM_HEADS], sh2[NUM_HEADS];
#pragma unroll
    for (int hh = 0; hh < NUM_HEADS; ++hh) {
        int bh = b * NUM_HEADS + hh;
        gv[hh] = gA[bh]; gm[hh] = gammaA[bh];
        sh0[hh] = shiftA[bh * 3 + 0]; sh1[hh] = shiftA[bh * 3 + 1]; sh2[hh] = shiftA[bh * 3 + 2];
    }

    // load scores, track per-head max
    float mx[NUM_HEADS] = {-3.4e38f, -3.4e38f, -3.4e38f, -3.4e38f};
    for (int i = 0; i < 8; ++i) {
        int n = i * 256 + t;
#pragma unroll
        for (int hh = 0; hh < NUM_HEADS; ++hh) {
            float s = score[((size_t)(b * NUM_HEADS + hh)) * N_ADDR + n];
            sbuf[hh * N_ADDR + n] = s;
            mx[hh] = fmaxf(mx[hh], s);
        }
    }
#pragma unroll
    for (int hh = 0; hh < NUM_HEADS; ++hh) red[hh * 256 + t] = mx[hh];
    __syncthreads();
    for (int off = 128; off > 0; off >>= 1) {
        if (t < off)
#pragma unroll
            for (int hh = 0; hh < NUM_HEADS; ++hh)
                red[hh * 256 + t] = fmaxf(red[hh * 256 + t], red[hh * 256 + t + off]);
        __syncthreads();
    }
    float gmax[NUM_HEADS];
#pragma unroll
    for (int hh = 0; hh < NUM_HEADS; ++hh) gmax[hh] = red[hh * 256];
    __syncthreads();

    // exp + sum
    float sm[NUM_HEADS] = {0, 0, 0, 0};
    for (int i = 0; i < 8; ++i) {
        int n = i * 256 + t;
#pragma unroll
        for (int hh = 0; hh < NUM_HEADS; ++hh) {
            float e = expf(sbuf[hh * N_ADDR + n] - gmax[hh]);
            sbuf[hh * N_ADDR + n] = e;
            sm[hh] += e;
        }
    }
#pragma unroll
    for (int hh = 0; hh < NUM_HEADS; ++hh) red[hh * 256 + t] = sm[hh];
    __syncthreads();
    for (int off = 128; off > 0; off >>= 1) {
        if (t < off)
#pragma unroll
            for (int hh = 0; hh < NUM_HEADS; ++hh)
                red[hh * 256 + t] += red[hh * 256 + t + off];
        __syncthreads();
    }
    float gsum[NUM_HEADS];
#pragma unroll
    for (int hh = 0; hh < NUM_HEADS; ++hh) gsum[hh] = 1.0f / red[hh * 256];
    __syncthreads();

    // wg = g*wc + (1-g)*wt_mix  (in place, own slots)
    for (int i = 0; i < 8; ++i) {
        int n = i * 256 + t;
#pragma unroll
        for (int hh = 0; hh < NUM_HEADS; ++hh) {
            float wc = sbuf[hh * N_ADDR + n] * gsum[hh];
            float wmx = wt_mix[((size_t)(b * NUM_HEADS + hh)) * N_ADDR + n];
            sbuf[hh * N_ADDR + n] = gv[hh] * wc + (1.0f - gv[hh]) * wmx;
        }
    }
    __syncthreads();

    // circular conv + pow(gamma) in registers; reduce sum
    float wp[NUM_HEADS][8];
    float ps[NUM_HEADS] = {0, 0, 0, 0};
    for (int i = 0; i < 8; ++i) {
        int n = i * 256 + t;
        int np = (n + 1) & (N_ADDR - 1);
        int nm = (n + N_ADDR - 1) & (N_ADDR - 1);
#pragma unroll
        for (int hh = 0; hh < NUM_HEADS; ++hh) {
            float wsv = sbuf[hh * N_ADDR + np] * sh0[hh]
                      + sbuf[hh * N_ADDR + n]  * sh1[hh]
                      + sbuf[hh * N_ADDR + nm] * sh2[hh];
            float p = expf(gm[hh] * logf(wsv + 1e-12f));
            wp[hh][i] = p;
            ps[hh] += p;
        }
    }
#pragma unroll
    for (int hh = 0; hh < NUM_HEADS; ++hh) red[hh * 256 + t] = ps[hh];
    __syncthreads();
    for (int off = 128; off > 0; off >>= 1) {
        if (t < off)
#pragma unroll
            for (int hh = 0; hh < NUM_HEADS; ++hh)
                red[hh * 256 + t] += red[hh * 256 + t + off];
        __syncthreads();
    }
    float pinv[NUM_HEADS];
#pragma unroll
    for (int hh = 0; hh < NUM_HEADS; ++hh) pinv[hh] = 1.0f / red[hh * 256];

    for (int i = 0; i < 8; ++i) {
        int n = i * 256 + t;
#pragma unroll
        for (int hh = 0; hh < NUM_HEADS; ++hh)
            wnew[((size_t)(b * NUM_HEADS + hh)) * N_ADDR + n] = wp[hh][i] * pinv[hh];
    }
}

// ---------------- K8: mem_new = mem * prod_h(1 - w*erase) + sum_h w*add ----------------
__global__ void memupdate_kernel(const float* __restrict__ mem, const float* __restrict__ wnew,
                                 const float* __restrict__ erase, const float* __restrict__ addv,
                                 float* __restrict__ mem_new) {
    int row = blockIdx.x * 4 + (threadIdx.x >> 6);   // b*N_ADDR + n
    int m = threadIdx.x & 63;
    int b = row >> 11;
    int n = row & (N_ADDR - 1);
    float keep = 1.0f, add = 0.0f;
#pragma unroll
    for (int hh = 0; hh < NUM_HEADS; ++hh) {
        float w = wnew[((size_t)(b * NUM_HEADS + hh)) * N_ADDR + n];
        keep *= (1.0f - w * erase[(b * NUM_HEADS + hh) * MSZ + m]);
        add += w * addv[(b * NUM_HEADS + hh) * MSZ + m];
    }
    size_t idx = ((size_t)b * N_ADDR + n) * MSZ + m;
    float mv = __builtin_nontemporal_load(mem + idx);          // last use of mem
    __builtin_nontemporal_store(mv * keep + add, mem_new + idx); // streaming output
}

// ---------------- launcher ----------------
extern "C" void kernel_launch(void* const* d_in, const int* in_sizes, int n_in,
                              void* d_out, int out_size, void* d_ws, size_t ws_size,
                              hipStream_t stream) {
    const float* tm_input = (const float*)d_in[0];
    const float* tm_state = (const float*)d_in[1];
    const float* wt       = (const float*)d_in[2];
    const float* mem      = (const float*)d_in[3];
    const float* wtd      = (const float*)d_in[4];
    const float* W_f = (const float*)d_in[5];  const float* b_f = (const float*)d_in[6];
    const float* W_h = (const float*)d_in[7];  const float* b_h = (const float*)d_in[8];
    const float* W_s = (const float*)d_in[9];  const float* b_s = (const float*)d_in[10];
    const float* W_o = (const float*)d_in[11]; const float* b_o = (const float*)d_in[12];
    const float* W_u = (const float*)d_in[13]; const float* b_u = (const float*)d_in[14];

    float* out = (float*)d_out;
    float* ws = (float*)d_ws;

    // output layout (floats)
    float* out_output = out;                        // B*OUT_DIM
    float* out_state  = out + 32768;                // B*STATE
    float* out_wtnew  = out + 98304;                // B*4*N_ADDR
    float* out_memnew = out + 1146880;              // B*N_ADDR*M
    float* out_wtdyn  = out + 17924096;             // B*4*N_ADDR

    // workspace layout (floats)
    float* ws_f     = ws;                  // 512
    float* ws_h     = ws + 512;            // 256
    float* ws_wtmix = ws + 768;            // 1048576
    float* ws_ctrl  = ws + 1049344;        // 131072
    float* ws_upd   = ws + 1180416;        // 101376
    float* ws_erase = ws + 1281792;        // 32768
    float* ws_addv  = ws + 1314560;        // 32768
    float* ws_key   = ws + 1347328;        // 32768
    float* ws_shift = ws + 1380096;        // 1536
    float* ws_gamma = ws + 1381632;        // 512
    float* ws_beta  = ws + 1382144;        // 512
    float* ws_g     = ws + 1382656;        // 512
    float* ws_kn    = ws + 1383168;        // 512
    float* ws_score = ws + 1383680;        // 1048576
    float* ws_part  = ws + 2432256;        // 262144

    gates_kernel<<<BSZ, 192, 0, stream>>>(tm_input, tm_state, W_f, b_f, W_h, b_h, ws_f, ws_h);

    int tot = BSZ * NUM_HEADS * N_ADDR;
    wtmix_kernel<<<(tot + 255) / 256, 256, 0, stream>>>(wt, wtd, ws_f, ws_h, ws_wtmix, out_wtdyn);

    concat_kernel<<<(BSZ * GATE_IN + 255) / 256, 256, 0, stream>>>(tm_input, tm_state, ws_ctrl);
    read_partial_kernel<<<BSZ * 8, 256, 0, stream>>>(ws_wtmix, mem, ws_part);
    read_reduce_kernel<<<(BSZ * 256 + 255) / 256, 256, 0, stream>>>(ws_part, ws_ctrl);

    // WMMA GEMMs on ctrl_in (128x1024)
    {
        int tilesS = 8 * ((STATE + 15) / 16);
        wmma_gemm_f32<<<(tilesS + 3) / 4, 128, 0, stream>>>(ws_ctrl, W_s, b_s, out_state,
                                                            BSZ, STATE, CTRL_IN, 1, tilesS);
        int tilesO = 8 * ((OUT_DIM + 15) / 16);
        wmma_gemm_f32<<<(tilesO + 3) / 4, 128, 0, stream>>>(ws_ctrl, W_o, b_o, out_output,
                                                            BSZ, OUT_DIM, CTRL_IN, 1, tilesO);
        int tilesU = 8 * ((UPDATE_SIZE + 15) / 16);
        wmma_gemm_f32<<<(tilesU + 3) / 4, 128, 0, stream>>>(ws_ctrl, W_u, b_u, ws_upd,
                                                            BSZ, UPDATE_SIZE, CTRL_IN, 0, tilesU);
    }

    iface_kernel<<<BSZ * NUM_HEADS, MSZ, 0, stream>>>(ws_upd, ws_erase, ws_addv, ws_key,
                                                      ws_shift, ws_gamma, ws_beta, ws_g, ws_kn);

    score_kernel<<<BSZ * 8, 256, 0, stream>>>(mem, ws_key, ws_beta, ws_kn, ws_score);

    address_kernel<<<BSZ, 256, 0, stream>>>(ws_score, ws_wtmix, ws_shift, ws_gamma,
                                            ws_g, out_wtnew);

    memupdate_kernel<<<(BSZ * N_ADDR) / 4, 256, 0, stream>>>(mem, out_wtnew, ws_erase,
                                                             ws_addv, out_memnew);
}